// STFTResonanceGenerator_36867999269157
// MI455X (gfx1250) — compile-verified
//
#include <hip/hip_runtime.h>
#include <math.h>

#define NSAMP 131072
#define WIN   2048
#define STEP  1024
#define NC    1025      // n_coeffs
#define NF    128       // n_frames
#define ZD    128       // z_dim
#define BATCH 64
#define NTF   131200    // NC * NF
#define IMP   4096

typedef __attribute__((ext_vector_type(2))) float v2f;
typedef __attribute__((ext_vector_type(8))) float v8f;

#define TWO_PI 6.28318530717958647692f

// ---------------------------------------------------------------------------
// Stockham radix-2 complex FFT of length 2048 in LDS (autosorting, no bitrev).
// sign = -1 : forward DFT (unscaled).  sign = +1 : inverse DFT (unscaled).
// Returns pointer to the buffer holding the result. All threads must call.
// ---------------------------------------------------------------------------
__device__ __forceinline__ float2* fft2048(float2* bufA, float2* bufB,
                                           float sign, int tid, int nt) {
  float2* src = bufA;
  float2* dst = bufB;
  int n = 2048;
  int ls = 0;                       // log2(stride)
  #pragma unroll 1
  for (int stage = 0; stage < 11; ++stage) {
    const int m = n >> 1;
    const int s = 1 << ls;
    const float theta0 = TWO_PI / (float)n;
    for (int idx = tid; idx < 1024; idx += nt) {
      const int p = idx >> ls;
      const int q = idx & (s - 1);
      float ss, cc;
      __builtin_sincosf(theta0 * (float)p, &ss, &cc);
      const float wr = cc;
      const float wi = sign * ss;
      const float2 a = src[q + (p << ls)];
      const float2 b = src[q + ((p + m) << ls)];
      float2 sum; sum.x = a.x + b.x; sum.y = a.y + b.y;
      float2 dif; dif.x = a.x - b.x; dif.y = a.y - b.y;
      float2 tw;  tw.x = dif.x * wr - dif.y * wi;
                  tw.y = dif.x * wi + dif.y * wr;
      dst[q + ((2 * p)     << ls)] = sum;
      dst[q + ((2 * p + 1) << ls)] = tw;
    }
    __syncthreads();
    float2* t = src; src = dst; dst = t;
    n >>= 1; ls += 1;
  }
  return src;
}

__device__ __forceinline__ float hann_win(int j) {
  return 0.5f - 0.5f * cosf(TWO_PI * (float)j / (float)WIN);
}

// ---------------------------------------------------------------------------
// Kernel 2: tf = squash(z @ W_tf + b_tf), f32 WMMA 16x16x4.
// One wave per 16-column tile of N=131200 (8200 tiles = 1025 blocks x 8 waves).
// Each wave computes all 4 M-tiles (M=64) so W_tf is streamed exactly once.
// tf layout: [batch][n] with n = c*128 + f (matches matmul output directly).
// ---------------------------------------------------------------------------
__global__ __launch_bounds__(256) void tf_matmul_kernel(
    const float* __restrict__ z, const float* __restrict__ W,
    const float* __restrict__ bias, float* __restrict__ tf) {
  const int wave = blockIdx.x * 8 + (threadIdx.x >> 5);   // 0..8199 (exact)
  const int lane = threadIdx.x & 31;
  const int half = lane >> 4;                             // 0: K pair {0,1}, 1: {2,3}
  const int r16  = lane & 15;
  const int col  = wave * 16 + r16;                       // global N column

  v8f acc0 = {}, acc1 = {}, acc2 = {}, acc3 = {};
  #pragma unroll 4
  for (int k0 = 0; k0 < ZD; k0 += 4) {
    const int kA = k0 + 2 * half;
    v2f bm;
    bm.x = W[(size_t)kA       * NTF + col];
    bm.y = W[(size_t)(kA + 1) * NTF + col];
    v2f a0, a1, a2, a3;
    a0.x = z[(r16     ) * ZD + kA]; a0.y = z[(r16     ) * ZD + kA + 1];
    a1.x = z[(r16 + 16) * ZD + kA]; a1.y = z[(r16 + 16) * ZD + kA + 1];
    a2.x = z[(r16 + 32) * ZD + kA]; a2.y = z[(r16 + 32) * ZD + kA + 1];
    a3.x = z[(r16 + 48) * ZD + kA]; a3.y = z[(r16 + 48) * ZD + kA + 1];
    acc0 = __builtin_amdgcn_wmma_f32_16x16x4_f32(false, a0, false, bm, (short)0, acc0, false, false);
    acc1 = __builtin_amdgcn_wmma_f32_16x16x4_f32(false, a1, false, bm, (short)0, acc1, false, false);
    acc2 = __builtin_amdgcn_wmma_f32_16x16x4_f32(false, a2, false, bm, (short)0, acc2, false, false);
    acc3 = __builtin_amdgcn_wmma_f32_16x16x4_f32(false, a3, false, bm, (short)0, acc3, false, false);
  }

  const float bc = bias[col];
  #pragma unroll
  for (int v = 0; v < 8; ++v) {
    const int rr = v + 8 * half;    // row within a 16-row M tile
    const float x0 = acc0[v] + bc;
    const float x1 = acc1[v] + bc;
    const float x2 = acc2[v] + bc;
    const float x3 = acc3[v] + bc;
    tf[(size_t)(rr     ) * NTF + col] = 0.02f + 0.9702f / (1.0f + expf(-x0));
    tf[(size_t)(rr + 16) * NTF + col] = 0.02f + 0.9702f / (1.0f + expf(-x1));
    tf[(size_t)(rr + 32) * NTF + col] = 0.02f + 0.9702f / (1.0f + expf(-x2));
    tf[(size_t)(rr + 48) * NTF + col] = 0.02f + 0.9702f / (1.0f + expf(-x3));
  }
}

// ---------------------------------------------------------------------------
// Kernel 3: cur_spec[b][f][k] = rfft(hann * impulse frame), frames 0..3 only
// (frames >= 4 are exactly zero: impulse is 4096 samples, hop 1024).
// One block per (b, f); complex FFT of real input, keep bins 0..1024.
// ---------------------------------------------------------------------------
__global__ __launch_bounds__(256) void rfft_frames_kernel(
    const float* __restrict__ imp, float2* __restrict__ cur) {
  __shared__ float2 bufA[2048];
  __shared__ float2 bufB[2048];
  const int b   = blockIdx.x >> 2;
  const int f   = blockIdx.x & 3;
  const int tid = threadIdx.x;
  for (int j = tid; j < WIN; j += 256) {
    const int si = f * STEP + j;
    const float x = (si < IMP) ? imp[(size_t)b * IMP + si] : 0.0f;
    bufA[j].x = x * hann_win(j);
    bufA[j].y = 0.0f;
  }
  __syncthreads();
  float2* r = fft2048(bufA, bufB, -1.0f, tid, 256);
  for (int k = tid; k < NC; k += 256) {
    cur[(size_t)blockIdx.x * NC + k] = r[k];
  }
}

// ---------------------------------------------------------------------------
// Kernel 4: per-bin spectral recurrence over 128 frames.
//   ps  = H(spec_{f-1})      (imag of bins 0 and 1024 dropped, = rfft∘irfft)
//   rot = (re*c + im*s, -(re*s + im*c)),  g = pi*k/1024
//   spec_f = (cur_f + rot) * tf_f
// 64*1025 = 65600 independent chains.
// ---------------------------------------------------------------------------
__global__ __launch_bounds__(256) void scan_kernel(
    const float2* __restrict__ cur, const float* __restrict__ tf,
    float2* __restrict__ spec) {
  const int t = blockIdx.x * 256 + threadIdx.x;
  if (t >= BATCH * NC) return;
  const int b = t / NC;
  const int k = t - b * NC;

  float sg, cg;
  __builtin_sincosf((3.14159265358979323846f / 1024.0f) * (float)k, &sg, &cg);
  const bool edge = (k == 0) || (k == NC - 1);

  const float*  tfb   = tf   + (size_t)b * NTF + (size_t)k * NF;  // tf[b][k*128+f]
  const float2* curb  = cur  + (size_t)b * 4 * NC + k;
  float2*       specb = spec + (size_t)b * NF * NC + k;

  float pre = 0.0f, pim = 0.0f;   // frame -1 audio is silence
  #pragma unroll 4
  for (int f = 0; f < NF; ++f) {
    const float tfi = tfb[f];
    const float reh = pre;
    const float imh = edge ? 0.0f : pim;
    const float rr  =  reh * cg + imh * sg;
    const float ri  = -(reh * sg + imh * cg);
    float crr = 0.0f, cri = 0.0f;
    if (f < 4) { const float2 cs = curb[(size_t)f * NC]; crr = cs.x; cri = cs.y; }
    const float re = (crr + rr) * tfi;
    const float im = (cri + ri) * tfi;
    specb[(size_t)f * NC] = make_float2(re, im);
    pre = re; pim = im;
  }
}

// ---------------------------------------------------------------------------
// Kernel 5: irfft(spec, 2048) * hann, overlap-add (hop 1024) via atomicAdd.
// Each output sample receives <= 2 float adds (commutative => deterministic
// given the preceding memset). One block per (b, f).
// ---------------------------------------------------------------------------
__global__ __launch_bounds__(256) void irfft_ola_kernel(
    const float2* __restrict__ spec, float* __restrict__ out) {
  __shared__ float2 bufA[2048];
  __shared__ float2 bufB[2048];
  const int bf  = blockIdx.x;          // b*128 + f
  const int b   = bf >> 7;
  const int f   = bf & 127;
  const int tid = threadIdx.x;
  const float2* sp = spec + (size_t)bf * NC;

  // Hermitian extension: full[k] = S[k] (k<=1024), conj(S[2048-k]) otherwise;
  // imag of DC and Nyquist ignored (pocketfft c2r semantics).
  for (int k = tid; k < WIN; k += 256) {
    const int kk = (k <= 1024) ? k : (WIN - k);
    float2 v = sp[kk];
    if (kk == 0 || kk == 1024) v.y = 0.0f;
    if (k > 1024) v.y = -v.y;
    bufA[k] = v;
  }
  __syncthreads();
  float2* r = fft2048(bufA, bufB, +1.0f, tid, 256);

  const float scale = 1.0f / 2048.0f;
  for (int j = tid; j < WIN; j += 256) {
    const float a = r[j].x * scale * hann_win(j);
    const int o = f * STEP + j;
    if (o < NSAMP) atomicAdd(&out[(size_t)b * NSAMP + o], a);
  }
}

// ---------------------------------------------------------------------------
extern "C" void kernel_launch(void* const* d_in, const int* in_sizes, int n_in,
                              void* d_out, int out_size, void* d_ws, size_t ws_size,
                              hipStream_t stream) {
  const float* z    = (const float*)d_in[0];   // (64,1,128)
  const float* imp  = (const float*)d_in[1];   // (64,1,4096)
  const float* W    = (const float*)d_in[2];   // (128, 131200)
  const float* bias = (const float*)d_in[3];   // (131200,)
  float* out = (float*)d_out;                  // (64,1,131072)

  // Workspace layout (all 8-byte aligned):
  //   tf   : BATCH*NTF floats        = 33,587,200 B
  //   cur  : BATCH*4*NC  float2      =  2,099,200 B
  //   spec : BATCH*NF*NC float2      = 67,174,400 B   (total ~103 MB)
  char* ws = (char*)d_ws;
  float*  tf   = (float*)ws;
  float2* cur  = (float2*)(ws + (size_t)BATCH * NTF * 4);
  float2* spec = (float2*)(ws + (size_t)BATCH * NTF * 4 + (size_t)BATCH * 4 * NC * 8);

  hipMemsetAsync(d_out, 0, (size_t)out_size * sizeof(float), stream);
  tf_matmul_kernel  <<<dim3(1025),          dim3(256), 0, stream>>>(z, W, bias, tf);
  rfft_frames_kernel<<<dim3(BATCH * 4),     dim3(256), 0, stream>>>(imp, cur);
  scan_kernel       <<<dim3((BATCH * NC + 255) / 256), dim3(256), 0, stream>>>(cur, tf, spec);
  irfft_ola_kernel  <<<dim3(BATCH * NF),    dim3(256), 0, stream>>>(spec, out);
}